// CTCCASLS_v3_2113123910197
// MI455X (gfx1250) — compile-verified
//
#include <hip/hip_runtime.h>

#ifndef __has_builtin
#define __has_builtin(x) 0
#endif

#if __has_builtin(__builtin_amdgcn_tensor_load_to_lds) && __has_builtin(__builtin_amdgcn_s_wait_tensorcnt)
#define HAVE_TDM 1
#endif
#if __has_builtin(__builtin_amdgcn_wmma_f32_16x16x4_f32)
#define HAVE_WMMA 1
#endif

#define Bsz  4096
#define Tt   200
#define Ccls 38
#define Ssz  30
#define Lext 61          // 2*S+1 lattice states
#define NEGV (-1e9f)

typedef float        v2f   __attribute__((ext_vector_type(2)));
typedef float        v8f   __attribute__((ext_vector_type(8)));
typedef unsigned int u32x4 __attribute__((ext_vector_type(4)));
typedef int          i32x4 __attribute__((ext_vector_type(4)));
typedef int          i32x8 __attribute__((ext_vector_type(8)));

static __device__ __forceinline__ float lae2(float a, float b) {
  float m = fmaxf(a, b);
  float d = fminf(a, b) - m;
  return m + __logf(1.0f + __expf(d));
}
static __device__ __forceinline__ float lae3(float a, float b, float c) {
  float m = fmaxf(fmaxf(a, b), c);
  return m + __logf(__expf(a - m) + __expf(b - m) + __expf(c - m));
}
static __device__ __forceinline__ float wred_sum(float v) {
  for (int m = 16; m > 0; m >>= 1) v += __shfl_xor(v, m);
  return v;
}
static __device__ __forceinline__ float wred_max(float v) {
  for (int m = 16; m > 0; m >>= 1) v = fmaxf(v, __shfl_xor(v, m));
  return v;
}

// ---------------------------------------------------------------------------
// CTC forward: one wave32 per sample. Lane k owns lattice states 2k, 2k+1.
// The [T x C] logit slab is DMA'd to LDS by the Tensor Data Mover; emissions
// emit[t,l] = lp[t, ext[l]] are produced 16 timesteps at a time on the matrix
// pipe as an exact one-hot GEMM with v_wmma_f32_16x16x4_f32.
// ---------------------------------------------------------------------------
__global__ __launch_bounds__(32) void ctc_kernel(const float* __restrict__ inp,
                                                 const int*   __restrict__ target,
                                                 const int*   __restrict__ ilen,
                                                 const int*   __restrict__ tlen,
                                                 float*       __restrict__ ws_ctc) {
  __shared__ float s_lp[Tt * Ccls];   // raw logits for this sample (30.4 KB)
  __shared__ float s_E[16 * 64];      // emission chunk: 16 timesteps x 64 states
  __shared__ float s_lse[Tt];         // per-row logsumexp
  __shared__ int   s_ext[64];         // blank-interleaved labels (padded)
  __shared__ float s_alpha[64];

  const int b    = blockIdx.x;
  const int lane = threadIdx.x;

  // ----- Stage logits [T x C] into LDS via the Tensor Data Mover -----
#ifdef HAVE_TDM
  {
    const unsigned long long gaddr =
        (unsigned long long)(const void*)(inp + (size_t)b * Tt * Ccls);
    u32x4 g0;
    g0[0] = 1u;                                                   // count=1, user D#
    g0[1] = (unsigned)(unsigned long long)(const void*)&s_lp[0];  // lds_addr = flat addr[31:0]
    g0[2] = (unsigned)(gaddr & 0xFFFFFFFFull);                    // global_addr lo
    g0[3] = (unsigned)((gaddr >> 32) & 0x01FFFFFFull) | (2u << 30); // addr[56:32] | type=2
    i32x8 g1;
    g1[0] = (2 << 16);          // workgroup_mask=0, data_size=2 (4 bytes)
    g1[1] = (Ccls << 16);       // tensor_dim0[15:0] = 38
    g1[2] = (Tt << 16);         // tensor_dim0 hi=0 | tensor_dim1[15:0] = 200
    g1[3] = (Ccls << 16);       // tensor_dim1 hi=0 | tile_dim0 = 38
    g1[4] = Tt;                 // tile_dim1 = 200, tile_dim2 = 0
    g1[5] = Ccls;               // tensor_dim0_stride lo32 = 38
    g1[6] = 0;
    g1[7] = 0;
    i32x4 gz4; gz4[0] = gz4[1] = gz4[2] = gz4[3] = 0;   // groups 2/3 unused (2-D tile)
    i32x8 gz8;
    gz8[0] = gz8[1] = gz8[2] = gz8[3] = 0;
    gz8[4] = gz8[5] = gz8[6] = gz8[7] = 0;
    __builtin_amdgcn_tensor_load_to_lds(g0, g1, gz4, gz4, gz8, 0);
  }
#else
  for (int i = lane; i < Tt * Ccls; i += 32)
    s_lp[i] = inp[(size_t)b * Tt * Ccls + i];
#endif

  // Extended (blank-interleaved) label sequence while the DMA is in flight.
  for (int n = lane; n < 64; n += 32) {
    int e = 0;
    if (n < Lext && (n & 1)) e = target[b * Ssz + (n >> 1)];
    s_ext[n] = e;
  }

#ifdef HAVE_TDM
  __builtin_amdgcn_s_wait_tensorcnt(0);
#endif
  __syncthreads();

  // Per-timestep logsumexp (log-softmax denominator), lanes strided over rows.
  for (int r = lane; r < Tt; r += 32) {
    const float* row = &s_lp[r * Ccls];
    float m = row[0];
#pragma unroll
    for (int c = 1; c < Ccls; ++c) m = fmaxf(m, row[c]);
    float s = 0.f;
#pragma unroll
    for (int c = 0; c < Ccls; ++c) s += __expf(row[c] - m);
    s_lse[r] = m + __logf(s);
  }
  __syncthreads();

  const int il = ilen[b];
  const int tl = tlen[b];

  // allow_skip for odd state 2k+1 (label s=k): label differs from previous label
  const int  labP   = (lane >= 1) ? s_ext[2 * lane - 1] : 0;
  const int  labC   = s_ext[(2 * lane + 1) & 63];
  const bool skipOK = (lane == 0) ? true : (labC != labP);
  const bool vE     = (lane <= 30);   // state 2k   valid (<=60)
  const bool vO     = (lane <= 29);   // state 2k+1 valid (<=59)

  float aE = (lane == 0) ? 0.f : NEGV;
  float aO = NEGV;

  // One-hot B-operand metadata (constant across chunks)
  const int extN0 = s_ext[(lane & 15) + 0];
  const int extN1 = s_ext[(lane & 15) + 16];
  const int extN2 = s_ext[(lane & 15) + 32];
  const int extN3 = s_ext[(lane & 15) + 48];
  const int koff  = (lane & 16) ? 2 : 0;   // K held by VGPR0 within a K=4 slice
  const int rowL  = lane & 15;             // M (A) / N (B,D) sub-index
  const int Mbase = (lane & 16) ? 8 : 0;   // C/D row base for this half-wave

  for (int chunk = 0; chunk < (Tt + 15) / 16; ++chunk) {
    const int t0 = chunk * 16;

#ifdef HAVE_WMMA
    // E(16x64) = LP(16x38, zero-padded K) x OneHot(38x64) on the matrix pipe.
    v8f acc0 = {0, 0, 0, 0, 0, 0, 0, 0};
    v8f acc1 = {0, 0, 0, 0, 0, 0, 0, 0};
    v8f acc2 = {0, 0, 0, 0, 0, 0, 0, 0};
    v8f acc3 = {0, 0, 0, 0, 0, 0, 0, 0};
    const int rowA = min(t0 + rowL, Tt - 1);
#pragma unroll
    for (int kk = 0; kk < 10; ++kk) {
      const int k0 = kk * 4 + koff;
      v2f a;
      a[0] = s_lp[rowA * Ccls + min(k0, Ccls - 1)];
      a[1] = s_lp[rowA * Ccls + min(k0 + 1, Ccls - 1)];
      // one-hot columns: exact 0/1 selectors, never match padded K >= 38
      v2f b0; b0[0] = (extN0 == k0) ? 1.f : 0.f; b0[1] = (extN0 == k0 + 1) ? 1.f : 0.f;
      v2f b1; b1[0] = (extN1 == k0) ? 1.f : 0.f; b1[1] = (extN1 == k0 + 1) ? 1.f : 0.f;
      v2f b2; b2[0] = (extN2 == k0) ? 1.f : 0.f; b2[1] = (extN2 == k0 + 1) ? 1.f : 0.f;
      v2f b3; b3[0] = (extN3 == k0) ? 1.f : 0.f; b3[1] = (extN3 == k0 + 1) ? 1.f : 0.f;
      acc0 = __builtin_amdgcn_wmma_f32_16x16x4_f32(false, a, false, b0, (short)0, acc0, false, false);
      acc1 = __builtin_amdgcn_wmma_f32_16x16x4_f32(false, a, false, b1, (short)0, acc1, false, false);
      acc2 = __builtin_amdgcn_wmma_f32_16x16x4_f32(false, a, false, b2, (short)0, acc2, false, false);
      acc3 = __builtin_amdgcn_wmma_f32_16x16x4_f32(false, a, false, b3, (short)0, acc3, false, false);
    }
#pragma unroll
    for (int j = 0; j < 8; ++j) {
      s_E[(Mbase + j) * 64 + rowL + 0]  = acc0[j];
      s_E[(Mbase + j) * 64 + rowL + 16] = acc1[j];
      s_E[(Mbase + j) * 64 + rowL + 32] = acc2[j];
      s_E[(Mbase + j) * 64 + rowL + 48] = acc3[j];
    }
#else
    for (int i = lane; i < 16 * 64; i += 32) {
      const int tloc = i >> 6, l = i & 63;
      s_E[i] = s_lp[min(t0 + tloc, Tt - 1) * Ccls + s_ext[l]];
    }
#endif
    __syncthreads();

    // Scan the 16 timesteps of this chunk (serial chain, 1 shuffle/step).
    const int tmax = min(16, Tt - t0);
    for (int tloc = 0; tloc < tmax; ++tloc) {
      const int   t     = t0 + tloc;
      const float lse_t = s_lse[t];
      const float eE = s_E[tloc * 64 + 2 * lane]     - lse_t;
      const float eO = s_E[tloc * 64 + 2 * lane + 1] - lse_t;
      float pO = __shfl_up(aO, 1);              // alpha[2k-1] from lane k-1
      pO = (lane == 0) ? NEGV : pO;
      const float nE = lae2(aE, pO) + eE;       // blank state: no skip path
      const float a2 = skipOK ? pO : NEGV;      // alpha[l-2] gated by allow_skip
      const float nO = lae3(aO, aE, a2) + eO;
      const bool  tv = (t < il);
      aE = (tv && vE) ? nE : (vE ? aE : NEGV);
      aO = (tv && vO) ? nO : (vO ? aO : NEGV);
    }
    __syncthreads();
  }

  if (vE) s_alpha[2 * lane] = aE;
  if (vO) s_alpha[2 * lane + 1] = aO;
  if (lane == 31) { s_alpha[61] = NEGV; s_alpha[62] = NEGV; s_alpha[63] = NEGV; }
  __syncthreads();

  if (lane == 0) {
    const int idx = 2 * tl;                         // tl in [5,30] -> idx in [10,60]
    float nll = -lae2(s_alpha[idx], s_alpha[idx - 1]);
    if (nll > 1e8f) nll = 0.f;                      // zero_infinity
    ws_ctc[b] = nll / (float)tl;
  }
}

// ---------------------------------------------------------------------------
// Aligned-KL label smoothing: one wave per sample; lanes parallel over C.
// ---------------------------------------------------------------------------
__global__ __launch_bounds__(32) void kl_kernel(const float* __restrict__ inp,
                                                const int*   __restrict__ target,
                                                const int*   __restrict__ tlen,
                                                const int*   __restrict__ pos,
                                                const float* __restrict__ ms,
                                                float*       __restrict__ ws_kld) {
  const int b    = blockIdx.x;
  const int lane = threadIdx.x;
  const int L    = tlen[b];
  const int c1   = lane;        // 0..31
  const int c2   = lane + 32;   // 32..63 (valid for lanes 0..5)

  float accum = 0.f;
  for (int s = 0; s < L; ++s) {
    const int p  = pos[b * Ssz + s];
    const int tg = target[b * Ssz + s];
    const int f  = (s == 0) ? (Ccls - 1) : target[b * Ssz + s - 1];

    const float* row = inp + ((size_t)b * Tt + p) * Ccls;
    const float x1 = row[c1];
    const float x2 = (c2 < Ccls) ? row[c2] : -1e30f;
    const float m  = wred_max(fmaxf(x1, x2));
    float se = __expf(x1 - m) + ((c2 < Ccls) ? __expf(x2 - m) : 0.f);
    se = wred_sum(se);
    const float lse = m + __logf(se);

    // SLS row: M[f, tg, c] = (c==0) ? 0 : ms[f-1, tg-1, c-1]   (f,tg in 1..37)
    const float* msrow = ms + ((size_t)(f - 1) * (Ccls - 1) + (size_t)(tg - 1)) * (Ccls - 1);
    const float s1 = (c1 == 0) ? 0.f : msrow[c1 - 1];
    const float s2 = (c2 < Ccls) ? msrow[c2 - 1] : 0.f;
    const float t1 = s1 + 1e-10f;
    const float t2 = s2 + 1e-10f;

    float kl = t1 * (__logf(t1) - (x1 - lse));
    if (c2 < Ccls) kl += t2 * (__logf(t2) - (x2 - lse));
    kl = wred_sum(kl);
    const float w = wred_sum(s1 + s2);
    accum += w * (kl / (float)Ccls);
  }

  if (lane == 0) {
    const float Lf = (float)L;
    const float smoothing = 1.f - __powf(0.95f, 1.f / Lf);
    ws_kld[b] = smoothing * (accum / Lf);
  }
}

// ---------------------------------------------------------------------------
// Deterministic fixed-order final reduction (no float atomics).
// ---------------------------------------------------------------------------
__global__ __launch_bounds__(256) void reduce_kernel(const float* __restrict__ ctc,
                                                     const float* __restrict__ kld,
                                                     float* __restrict__ out) {
  __shared__ float sc[256];
  __shared__ float sk[256];
  const int tid = threadIdx.x;
  float s1 = 0.f, s2 = 0.f;
  for (int i = tid; i < Bsz; i += 256) { s1 += ctc[i]; s2 += kld[i]; }
  sc[tid] = s1; sk[tid] = s2;
  __syncthreads();
  for (int st = 128; st > 0; st >>= 1) {
    if (tid < st) { sc[tid] += sc[tid + st]; sk[tid] += sk[tid + st]; }
    __syncthreads();
  }
  if (tid == 0) out[0] = sc[0] / (float)Bsz + sk[0];
}

extern "C" void kernel_launch(void* const* d_in, const int* in_sizes, int n_in,
                              void* d_out, int out_size, void* d_ws, size_t ws_size,
                              hipStream_t stream) {
  const float* inp    = (const float*)d_in[0];
  const int*   target = (const int*)d_in[1];
  const int*   ilen   = (const int*)d_in[2];
  const int*   tlen   = (const int*)d_in[3];
  const int*   pos    = (const int*)d_in[4];
  const float* ms     = (const float*)d_in[5];
  float*       out    = (float*)d_out;

  float* ws_ctc = (float*)d_ws;        // [B] per-sample nll/len
  float* ws_kld = ws_ctc + Bsz;        // [B] per-sample smoothing*per_sample

  ctc_kernel<<<Bsz, 32, 0, stream>>>(inp, target, ilen, tlen, ws_ctc);
  kl_kernel<<<Bsz, 32, 0, stream>>>(inp, target, tlen, pos, ms, ws_kld);
  reduce_kernel<<<1, 256, 0, stream>>>(ws_ctc, ws_kld, out);
}